// ESRNN_59931973648853
// MI455X (gfx1250) — compile-verified
//
#include <hip/hip_runtime.h>
#include <cstdint>
#include <cstddef>

// ES-RNN (Holt-Winters) scan for MI455X / gfx1250.
//  - 64 series per block, 64 blocks -> spread across WGPs (latency-bound chain)
//  - x tiles staged global->LDS by the Tensor Data Mover (one DMA per tile,
//    TENSORcnt + s_wait_tensorcnt), double-buffered; LDS row stride padded to
//    33 floats *by the TDM pad feature* -> bank-conflict-free compute reads
//  - outputs staged in LDS, written back fully coalesced
//  - seasonal ring in LDS (bank-conflict-free), l/b in VGPRs
//  - divisions via v_rcp_f32 + 1 Newton step (short critical chain)

namespace {
constexpr int kB    = 4096;
constexpr int kT    = 2048;
constexpr int kP    = 12;
constexpr int ROWS  = 64;           // series per block
constexpr int TILE  = 32;           // timesteps per LDS tile (kT % TILE == 0)
constexpr int XSTR  = TILE + 1;     // padded x row stride (33): gcd(33,64)=1
constexpr int OSTR  = ROWS + 1;     // padded out stride (65):  gcd(65,64)=1
constexpr int NTILE = kT / TILE;    // 64
}

typedef uint32_t u32x4 __attribute__((ext_vector_type(4)));
typedef int      i32x8 __attribute__((ext_vector_type(8)));
typedef int      i32x4 __attribute__((ext_vector_type(4)));

__device__ __forceinline__ uint32_t lds_addr32(const void* p) {
  // generic pointers into LDS carry the LDS byte offset in their low 32 bits
  return (uint32_t)(uintptr_t)p;
}

#if __has_builtin(__builtin_amdgcn_tensor_load_to_lds)
#define USE_TDM 1
// One TDM descriptor: 2D tile (tile_dim1=64 rows x tile_dim0=32 floats) from a
// row-major tensor with element stride 2048; LDS gets +1 dword pad per 32
// dwords -> effective LDS row stride = 33 floats.
__device__ __forceinline__ void tdm_load_tile(uint32_t lds_off, const float* gsrc) {
  const uint64_t ga = (uint64_t)(uintptr_t)gsrc;
  u32x4 g0;
  g0[0] = 1u;                                            // count=1, user mode
  g0[1] = lds_off;                                       // lds_addr (bytes)
  g0[2] = (uint32_t)ga;                                  // global_addr[31:0]
  g0[3] = (uint32_t)((ga >> 32) & 0x01FFFFFFu)           // global_addr[56:32]
        | (2u << 30);                                    // type=2 ("image")
  i32x8 g1;
  g1[0] = (int)((2u << 16)      // data_size = 4 bytes
              | (1u << 20)      // pad_enable
              | (4u << 22));    // pad_interval code 4 = 32 dwords (+pad_amount 0 = 1 dword)
  g1[1] = (int)((uint32_t)TILE << 16);   // tensor_dim0[15:0] in bits 63:48
  g1[2] = (int)((uint32_t)ROWS << 16);   // tensor_dim0 hi=0 ; tensor_dim1 = 64
  g1[3] = (int)((uint32_t)TILE << 16);   // tensor_dim1 hi=0 ; tile_dim0 = 32
  g1[4] = ROWS;                          // tile_dim1 = 64 ; tile_dim2 = 0
  g1[5] = kT;                            // tensor_dim0_stride = 2048 elements
  g1[6] = 0;                             // stride hi ; tensor_dim1_stride lo
  g1[7] = 0;
  i32x4 g2 = {1, 0, 0, 0};               // tensor_dim2 = 1 (benign)
  i32x4 g3 = {0, 1 << 16, 0, 0};         // tensor_dim4 = 1 (benign)
  i32x8 g4 = {0, 0, 0, 0, 0, 0, 0, 0};   // 6-arg toolchain form: zero-filled
  __builtin_amdgcn_tensor_load_to_lds(g0, g1, g2, g3, g4, 0);
}
#else
#define USE_TDM 0
__device__ __forceinline__ void async_copy_b32(uint32_t lds_off, const float* g) {
  asm volatile("global_load_async_to_lds_b32 %0, %1, off"
               :: "v"(lds_off), "v"(g) : "memory");
}
#endif

__device__ __forceinline__ float fast_rcp(float a) {
  float r = __builtin_amdgcn_rcpf(a);           // v_rcp_f32 (~1 ulp)
  float e = __builtin_fmaf(-a, r, 1.0f);        // one Newton step -> ~0.5 ulp
  return __builtin_fmaf(r, e, r);
}

__global__ __launch_bounds__(ROWS) void esrnn_scan(
    const float* __restrict__ x,  const float* __restrict__ l0,
    const float* __restrict__ b0, const float* __restrict__ s0,
    const float* __restrict__ pA, const float* __restrict__ pBt,
    const float* __restrict__ pPhi, const float* __restrict__ pG,
    float* __restrict__ out) {
  __shared__ float xbuf[2][ROWS * XSTR];   // staged x, [row][33] (TDM-padded)
  __shared__ float obuf[TILE * OSTR];      // staged out, [t][65]
  __shared__ float sbuf[kP * ROWS];        // seasonal ring, [period][row]

  const int tid  = threadIdx.x;
  const int row0 = blockIdx.x * ROWS;
  const int row  = row0 + tid;

  const float alpha = pA[0], beta = pBt[0], phi = pPhi[0], gamma = pG[0];
  const float oma = 1.0f - alpha;
  const float cb  = (1.0f - beta) * phi;   // (1-beta)*phi
  const float omg = 1.0f - gamma;

  float l = l0[row];
  float b = b0[row];
#pragma unroll
  for (int j = 0; j < kP; ++j)             // own-lane only; no barrier needed
    sbuf[j * ROWS + tid] = s0[row * kP + j];

  const uint32_t xoff0 = __builtin_amdgcn_readfirstlane(lds_addr32(&xbuf[0][0]));
  const uint32_t xoff1 = __builtin_amdgcn_readfirstlane(lds_addr32(&xbuf[1][0]));

  auto stage = [&](int tile, int buf) {
    const int tb = tile * TILE;
#if USE_TDM
    if (tid < 32)                          // one wave issues the DMA (TDM ignores EXEC)
      tdm_load_tile(buf ? xoff1 : xoff0, x + (size_t)row0 * kT + tb);
#else
#pragma unroll
    for (int j = 0; j < TILE; ++j) {
      const int f  = tid + ROWS * j;       // 0..2047
      const int r  = f >> 5;               // row within block
      const int tl = f & 31;               // t within tile
      async_copy_b32(lds_addr32(&xbuf[buf][r * XSTR + tl]),
                     x + (size_t)(row0 + r) * kT + tb + tl);
    }
#endif
  };

  auto wait_tiles = [&](bool more) {
#if USE_TDM
    if (more) __builtin_amdgcn_s_wait_tensorcnt((short)1);   // in-order: older DMA done
    else      __builtin_amdgcn_s_wait_tensorcnt((short)0);
#else
    if (more) asm volatile("s_wait_asynccnt 0x20" ::: "memory");
    else      asm volatile("s_wait_asynccnt 0x0"  ::: "memory");
#endif
  };

  stage(0, 0);

  int head = 0;                            // seasonal ring head (== s[0])
  for (int tile = 0; tile < NTILE; ++tile) {
    const int  cur  = tile & 1;
    const bool more = (tile + 1) < NTILE;
    if (more) stage(tile + 1, cur ^ 1);    // prefetch next tile (double buffer)
    wait_tiles(more);                      // this tile's DMA has landed
    __syncthreads();

    const float* xt = &xbuf[cur][tid * XSTR];  // this lane's row, stride-33 base
#pragma unroll 4
    for (int k = 0; k < TILE; ++k) {
      const int   h1   = (head + 1 == kP) ? 0 : head + 1;
      const float sf   = sbuf[head * ROWS + tid];   // s[0]  (12 steps old)
      const float ss   = sbuf[h1   * ROWS + tid];   // s[1]
      const float xv   = xt[k];
      const float rsf  = fast_rcp(sf);              // off the l/b critical chain
      const float lb   = __builtin_fmaf(phi, b, l);           // l + phi*b
      const float lnew = __builtin_fmaf(alpha * xv, rsf, oma * lb);
      const float bnew = __builtin_fmaf(beta, lnew - l, cb * b);
      const float d    = __builtin_fmaf(phi, bnew, lnew);     // l + b*phi
      const float rd   = fast_rcp(d);
      const float snew = __builtin_fmaf(gamma * xv, rd, omg * sf);
      obuf[k * OSTR + tid]       = d * ss;          // out_t
      sbuf[head * ROWS + tid]    = snew;            // rotate ring
      l = lnew; b = bnew; head = h1;
    }
    __syncthreads();

    // Coalesced write-back of the tile (128B per wave per store).
    const int tb = tile * TILE;
#pragma unroll
    for (int j = 0; j < TILE; ++j) {
      const int f  = tid + ROWS * j;
      const int r  = f >> 5;
      const int tl = f & 31;
      out[(size_t)(row0 + r) * kT + tb + tl] = obuf[tl * OSTR + r];
    }
    __syncthreads();   // obuf reads done before next tile's compute rewrites it
  }
}

extern "C" void kernel_launch(void* const* d_in, const int* /*in_sizes*/, int /*n_in*/,
                              void* d_out, int /*out_size*/, void* /*d_ws*/, size_t /*ws_size*/,
                              hipStream_t stream) {
  const float* x     = (const float*)d_in[0];
  const float* l0    = (const float*)d_in[1];
  const float* b0    = (const float*)d_in[2];
  const float* s0    = (const float*)d_in[3];
  const float* alpha = (const float*)d_in[4];
  const float* beta  = (const float*)d_in[5];
  const float* phi   = (const float*)d_in[6];
  const float* gamma = (const float*)d_in[7];
  float* out = (float*)d_out;

  esrnn_scan<<<kB / ROWS, ROWS, 0, stream>>>(x, l0, b0, s0, alpha, beta, phi, gamma, out);
}